// YOLO_38543036514446
// MI455X (gfx1250) — compile-verified
//
#include <hip/hip_runtime.h>
#include <hip/hip_bf16.h>
#include <math.h>

// ---------------------------------------------------------------------------
// YOLO-style detection post-process for MI455X (gfx1250, wave32).
//   B=8 images, A=49104 anchors, C=81 classes, TOP_K=1000, IoU 0.5.
// Pipeline: score/argmax -> exact top-k (radix select + bitonic) -> decode ->
//           WMMA-assisted IoU suppression-mask build -> sequential NMS -> out.
// d_out layout (floats): boxes[8*1000*4] | scores[8*1000] | cls[8*1000]
// ---------------------------------------------------------------------------

#define NIMG   8
#define NANCH  49104
#define NCLS   81
#define KTOP   1000
#define NTILE  63            // ceil(1000/16)
#define IOU_THR 0.5f
#define SCORE_THR 0.05f
#define BBOX_CLIP 4.1351666f // log(1000/16)

typedef __attribute__((ext_vector_type(2))) float v2f;
typedef __attribute__((ext_vector_type(8))) float v8f;

// monotone float -> unsigned key (order preserving)
__device__ __forceinline__ unsigned int flipf(float f) {
    unsigned int u = __float_as_uint(f);
    return u ^ ((u >> 31) ? 0xFFFFFFFFu : 0x80000000u);
}
__device__ __forceinline__ float unflipf(unsigned int u) {
    unsigned int v = u ^ ((u & 0x80000000u) ? 0x80000000u : 0xFFFFFFFFu);
    return __uint_as_float(v);
}

// ---------------------------------------------------------------------------
// Kernel 1: wave-per-anchor max/argmax over 81 logits.
// sigmoid is monotone: score = sigmoid(max logit); argmax(prob)==argmax(logit).
// Streams the 127MB labels tensor coalesced (lane-contiguous class reads).
// ---------------------------------------------------------------------------
__global__ void score_kernel(const float* __restrict__ labels,
                             float* __restrict__ cand,
                             int* __restrict__ clsIdx) {
    int wid  = (blockIdx.x * blockDim.x + threadIdx.x) >> 5;   // anchor id (global)
    int lane = threadIdx.x & 31;
    if (wid >= NIMG * NANCH) return;
    const float* row = labels + (size_t)wid * NCLS;
    float v = -INFINITY;
    int   ci = 0;
    for (int c = lane; c < NCLS; c += 32) {
        float x = row[c];
        if (x > v) { v = x; ci = c; }      // first-max keeps lowest index
    }
    for (int off = 16; off; off >>= 1) {
        float ov = __shfl_xor(v, off, 32);
        int   oi = __shfl_xor(ci, off, 32);
        if (ov > v || (ov == v && oi < ci)) { v = ov; ci = oi; }
    }
    if (lane == 0) {
        float score = 1.0f / (1.0f + __expf(-v));
        bool valid = (ci != 0) && (score > SCORE_THR);
        cand[wid]   = valid ? score : -1.0f;
        clsIdx[wid] = ci;
    }
}

// ---------------------------------------------------------------------------
// Kernel 2: exact top-1000 per image. One 256-thread block per image.
// MSB-first 8-bit radix select finds the pivot key; strictly-greater keys are
// compacted unordered; equal-to-pivot keys are taken in ascending-index order
// (chunked wave ballots) -> matches jax.lax.top_k tie behavior. A 1024-entry
// bitonic sort on (key<<32)|~idx yields the final sorted order.
// ---------------------------------------------------------------------------
__global__ void topk_kernel(const float* __restrict__ cand,
                            const int* __restrict__ clsIdx,
                            float* __restrict__ topv,
                            int* __restrict__ topi,
                            int* __restrict__ clsSel) {
    const int img = blockIdx.x;
    const float* c = cand + (size_t)img * NANCH;
    __shared__ unsigned int scnt[256];
    __shared__ unsigned long long sel[1024];
    __shared__ unsigned int sPivot, sMask, sRem, sGcnt, sTaken, sTotal, sDone;
    __shared__ unsigned int wcnt[8], woff[8];

    const int t = threadIdx.x, lane = t & 31, wv = t >> 5;

    if (t == 0) { sPivot = 0; sMask = 0; sRem = KTOP; sGcnt = 0; sTaken = 0; sDone = 0; }
    for (int i = t; i < 1024; i += 256) sel[i] = 0ull;
    __syncthreads();

    // ---- radix select (4 rounds of 8 bits, MSB first) ----
    for (int s = 24; s >= 0; s -= 8) {
        scnt[t] = 0;
        __syncthreads();
        unsigned int hm = sMask, ph = sPivot;
        for (int i = t; i < NANCH; i += 256) {
            unsigned int k = flipf(c[i]);
            if ((k & hm) == ph) atomicAdd(&scnt[(k >> s) & 255u], 1u);
        }
        __syncthreads();
        if (t == 0) {
            unsigned int rem = sRem, acc = 0;
            int d = 255;
            for (; d > 0; --d) {
                if (acc + scnt[d] >= rem) break;
                acc += scnt[d];
            }
            sRem   = rem - acc;
            sPivot = ph | ((unsigned int)d << s);
            sMask  = hm | (255u << s);
        }
        __syncthreads();
    }
    const unsigned int pivot = sPivot;
    const unsigned int rem = sRem;               // #equal-pivot entries to take
    const unsigned int gTarget = KTOP - rem;     // #strictly greater entries

    // ---- compact strictly-greater keys (order fixed by final sort) ----
    for (int i = t; i < NANCH; i += 256) {
        unsigned int k = flipf(c[i]);
        if (k > pivot) {
            unsigned int p = atomicAdd(&sGcnt, 1u);
            if (p < 1024u)
                sel[p] = ((unsigned long long)k << 32) | (unsigned int)(~i);
        }
    }
    __syncthreads();

    // ---- equal-to-pivot keys: take `rem` in ascending index order ----
    for (int base = 0; base < NANCH; base += 256) {
        if (sDone) break;
        int i = base + t;
        bool flag = (i < NANCH) && (flipf(c[i]) == pivot);
        unsigned long long bal = __ballot(flag);
        unsigned int m = (unsigned int)bal;      // wave32
        if (lane == 0) wcnt[wv] = __popc(m);
        __syncthreads();
        if (t == 0) {
            unsigned int tot = 0;
            for (int w = 0; w < 8; ++w) { woff[w] = tot; tot += wcnt[w]; }
            sTotal = tot;
        }
        __syncthreads();
        if (flag) {
            unsigned int slot = sTaken + woff[wv] + __popc(m & ((1u << lane) - 1u));
            if (slot < rem)
                sel[gTarget + slot] =
                    ((unsigned long long)pivot << 32) | (unsigned int)(~i);
        }
        __syncthreads();
        if (t == 0) {
            sTaken += sTotal;
            if (sTaken >= rem) sDone = 1;
        }
        __syncthreads();
    }

    // ---- bitonic sort 1024 descending (pads == 0 sort last) ----
    for (int ksz = 2; ksz <= 1024; ksz <<= 1) {
        for (int j = ksz >> 1; j > 0; j >>= 1) {
            for (int e = t; e < 1024; e += 256) {
                int p = e ^ j;
                if (p > e) {
                    bool desc = ((e & ksz) == 0);
                    unsigned long long a = sel[e], b = sel[p];
                    if ((a < b) == desc) { sel[e] = b; sel[p] = a; }
                }
            }
            __syncthreads();
        }
    }

    for (int s2 = t; s2 < KTOP; s2 += 256) {
        unsigned long long v = sel[s2];
        unsigned int key = (unsigned int)(v >> 32);
        int idx = (int)(~(unsigned int)v);
        topv[img * KTOP + s2] = unflipf(key);
        topi[img * KTOP + s2] = idx;
        clsSel[img * KTOP + s2] = clsIdx[(size_t)img * NANCH + idx];
    }
}

// ---------------------------------------------------------------------------
// Kernel 3: decode selected boxes + areas.
// ---------------------------------------------------------------------------
__global__ void decode_kernel(const float* __restrict__ rel,
                              const float* __restrict__ anc,
                              const int* __restrict__ topi,
                              float* __restrict__ boxes,
                              float* __restrict__ areas) {
    int g = blockIdx.x * blockDim.x + threadIdx.x;
    if (g >= NIMG * KTOP) return;
    int img = g / KTOP;
    int a = topi[g];
    const float* r  = rel + ((size_t)img * NANCH + a) * 4;
    const float* an = anc + ((size_t)img * NANCH + a) * 4;
    float w  = an[2] - an[0], h = an[3] - an[1];
    float cx = an[0] + 0.5f * w, cy = an[1] + 0.5f * h;
    float dx = r[0], dy = r[1];
    float dw = fminf(r[2], BBOX_CLIP), dh = fminf(r[3], BBOX_CLIP);
    float pcx = dx * w + cx, pcy = dy * h + cy;
    float pw = __expf(dw) * w, ph = __expf(dh) * h;
    float x1 = pcx - 0.5f * pw, y1 = pcy - 0.5f * ph;
    float x2 = pcx + 0.5f * pw, y2 = pcy + 0.5f * ph;
    float* b = boxes + (size_t)g * 4;
    b[0] = x1; b[1] = y1; b[2] = x2; b[3] = y2;
    areas[g] = fmaxf(x2 - x1, 0.0f) * fmaxf(y2 - y1, 0.0f);
}

// ---------------------------------------------------------------------------
// Kernel 4: suppression-mask build, one wave per 16x16 (i,j) tile.
// The union base term (area_i + area_j) is a rank-2 outer sum computed by one
// V_WMMA_F32_16X16X4_F32:  A row m = [area_i[m], 1, 0, 0],
//                          B col n = [1, area_j[n], 0, 0]^T.
// Per ISA f32 layouts: lanes 0-15 hold K=0,1 (VGPR0,1); lanes 16-31 hold K=2,3.
// Intersection min/max is VALU; bits (iou>thr && j>i) gathered via ballots.
// The IoU predicate uses the divide-free equivalent form:
//   inter/max(uni,eps) > thr  <=>  inter > thr*max(uni,eps)   (max(...) > 0)
// which avoids the IEEE v_div_scale/v_rcp/v_div_fmas/v_div_fixup sequence.
// ---------------------------------------------------------------------------
__global__ void supmask_kernel(const float* __restrict__ boxes,
                               const float* __restrict__ areas,
                               unsigned int* __restrict__ sup) {
    const int ti = blockIdx.x, tj = blockIdx.y, img = blockIdx.z;
    if (tj < ti) return;                         // only j >= i tiles matter
    __shared__ float bx[2][16][4];
    __shared__ float ar[2][16];
    const int lane = threadIdx.x;                // 32 threads = 1 wave

    {   // load the two 16-box tiles (lanes 0-15: i side, 16-31: j side)
        int side = lane >> 4;
        int m = lane & 15;
        int g = (side == 0 ? ti : tj) * 16 + m;
        if (g < KTOP) {
            const float* b = boxes + ((size_t)img * KTOP + g) * 4;
            bx[side][m][0] = b[0]; bx[side][m][1] = b[1];
            bx[side][m][2] = b[2]; bx[side][m][3] = b[3];
            ar[side][m] = areas[img * KTOP + g];
        } else {
            bx[side][m][0] = 0.f; bx[side][m][1] = 0.f;
            bx[side][m][2] = 0.f; bx[side][m][3] = 0.f;
            ar[side][m] = 0.f;
        }
    }
    __syncthreads();

    v2f a, b;
    if (lane < 16) {                 // K=0,1 halves
        a = (v2f){ar[0][lane], 1.0f};      // A[m][0]=area_i, A[m][1]=1
        b = (v2f){1.0f, ar[1][lane]};      // B[0][n]=1,      B[1][n]=area_j
    } else {                          // K=2,3 halves: zero padding
        a = (v2f){0.0f, 0.0f};
        b = (v2f){0.0f, 0.0f};
    }
    v8f cacc = {};
    // D[m][n] = area_i[m] + area_j[n]
    v8f d = __builtin_amdgcn_wmma_f32_16x16x4_f32(
        false, a, false, b, (short)0, cacc, false, false);

    const int n = lane & 15, half = lane >> 4;
    const float jx1 = bx[1][n][0], jy1 = bx[1][n][1];
    const float jx2 = bx[1][n][2], jy2 = bx[1][n][3];
    const unsigned int word  = (unsigned int)(tj * 16) >> 5;
    const unsigned int shift = (unsigned int)(tj * 16) & 31;
    unsigned int* srow = sup + (size_t)img * KTOP * 32;

#pragma unroll
    for (int r = 0; r < 8; ++r) {
        int m = r + half * 8;                    // D layout: VGPR r -> rows r, r+8
        float ix1 = bx[0][m][0], iy1 = bx[0][m][1];
        float ix2 = bx[0][m][2], iy2 = bx[0][m][3];
        float w2 = fmaxf(fminf(ix2, jx2) - fmaxf(ix1, jx1), 0.0f);
        float h2 = fmaxf(fminf(iy2, jy2) - fmaxf(iy1, jy1), 0.0f);
        float inter = w2 * h2;
        float uni = d[r] - inter;                // area_i + area_j - inter
        // divide-free IoU threshold test (exact: max(uni,eps) > 0)
        bool over = inter > IOU_THR * fmaxf(uni, 1e-9f);
        int ig = ti * 16 + m, jg = tj * 16 + n;
        bool pred = over && (jg > ig) && (jg < KTOP);
        unsigned long long bal = __ballot(pred);
        if (lane == 0) {
            unsigned int lo = (unsigned int)bal & 0xFFFFu;          // row ti*16+r
            unsigned int hi = ((unsigned int)bal >> 16) & 0xFFFFu;  // row ti*16+r+8
            int rowLo = ti * 16 + r, rowHi = rowLo + 8;
            if (lo && rowLo < KTOP) atomicOr(&srow[rowLo * 32 + word], lo << shift);
            if (hi && rowHi < KTOP) atomicOr(&srow[rowHi * 32 + word], hi << shift);
        }
    }
}

// ---------------------------------------------------------------------------
// Kernel 5: sequential NMS keep-propagation + masked outputs. One wave/image.
// Lane w owns keep bits [32w, 32w+32). keep[i] broadcast via __shfl (uniform).
// Matches: for i: sup = (iou[i]>thr)&(j>i)&keep[i]; keep &= ~sup.
// ---------------------------------------------------------------------------
__global__ void nms_out_kernel(const float* __restrict__ topv,
                               const int* __restrict__ clsSel,
                               const float* __restrict__ boxes,
                               const unsigned int* __restrict__ sup,
                               float* __restrict__ out) {
    const int img = blockIdx.x, lane = threadIdx.x;   // 32 threads = 1 wave
    const float* tv = topv + img * KTOP;
    const unsigned int* srow = sup + (size_t)img * KTOP * 32;

    unsigned int keepW = 0;
    for (int bit = 0; bit < 32; ++bit) {
        int i = lane * 32 + bit;
        if (i < KTOP && tv[i] > 0.0f) keepW |= 1u << bit;
    }
    for (int i = 0; i < KTOP; ++i) {
        unsigned int cur = __shfl(keepW, i >> 5, 32);     // uniform broadcast
        if ((cur >> (i & 31)) & 1u)
            keepW &= ~srow[(size_t)i * 32 + lane];
    }

    __shared__ unsigned int sk[32];
    sk[lane] = keepW;
    __syncthreads();

    float* outBoxes  = out;                       // [8,1000,4]
    float* outScores = out + NIMG * KTOP * 4;     // [8,1000]
    float* outCls    = out + NIMG * KTOP * 5;     // [8,1000] (written as float)
    for (int s = lane; s < KTOP; s += 32) {
        unsigned int k = (sk[s >> 5] >> (s & 31)) & 1u;
        const float* bb = boxes + ((size_t)img * KTOP + s) * 4;
        float* ob = outBoxes + ((size_t)img * KTOP + s) * 4;
        ob[0] = k ? bb[0] : 0.0f;
        ob[1] = k ? bb[1] : 0.0f;
        ob[2] = k ? bb[2] : 0.0f;
        ob[3] = k ? bb[3] : 0.0f;
        outScores[img * KTOP + s] = k ? tv[s] : 0.0f;
        outCls[img * KTOP + s]    = k ? (float)clsSel[img * KTOP + s] : 0.0f;
    }
}

// ---------------------------------------------------------------------------
extern "C" void kernel_launch(void* const* d_in, const int* in_sizes, int n_in,
                              void* d_out, int out_size, void* d_ws, size_t ws_size,
                              hipStream_t stream) {
    const float* labels = (const float*)d_in[0];   // [8,49104,81]
    const float* relb   = (const float*)d_in[1];   // [8,49104,4]
    const float* anch   = (const float*)d_in[2];   // [8,49104,4]
    float* out = (float*)d_out;

    // workspace carve (aligned to 256B)
    char* w = (char*)d_ws;
    size_t off = 0;
    auto carve = [&](size_t bytes) {
        void* p = w + off;
        off += (bytes + 255) & ~(size_t)255;
        return p;
    };
    float*        cand   = (float*)carve((size_t)NIMG * NANCH * 4);
    int*          clsIdx = (int*)  carve((size_t)NIMG * NANCH * 4);
    float*        topv   = (float*)carve((size_t)NIMG * KTOP * 4);
    int*          topi   = (int*)  carve((size_t)NIMG * KTOP * 4);
    int*          clsSel = (int*)  carve((size_t)NIMG * KTOP * 4);
    float*        boxes  = (float*)carve((size_t)NIMG * KTOP * 4 * 4);
    float*        areas  = (float*)carve((size_t)NIMG * KTOP * 4);
    unsigned int* sup    = (unsigned int*)carve((size_t)NIMG * KTOP * 32 * 4);

    hipMemsetAsync(sup, 0, (size_t)NIMG * KTOP * 32 * 4, stream);

    // 1: scores (8*49104 anchors, 1 wave each, 8 waves/block)
    score_kernel<<<NIMG * NANCH / 8, 256, 0, stream>>>(labels, cand, clsIdx);
    // 2: exact sorted top-1000 per image
    topk_kernel<<<NIMG, 256, 0, stream>>>(cand, clsIdx, topv, topi, clsSel);
    // 3: decode
    decode_kernel<<<(NIMG * KTOP + 255) / 256, 256, 0, stream>>>(relb, anch, topi,
                                                                 boxes, areas);
    // 4: WMMA suppression masks (63x63 tiles per image, 1 wave each)
    dim3 g4(NTILE, NTILE, NIMG);
    supmask_kernel<<<g4, 32, 0, stream>>>(boxes, areas, sup);
    // 5: sequential NMS + outputs (1 wave per image)
    nms_out_kernel<<<NIMG, 32, 0, stream>>>(topv, clsSel, boxes, sup, out);
}